// PolyIOUloss_41686952575033
// MI455X (gfx1250) — compile-verified
//
#include <hip/hip_runtime.h>
#include <math.h>

// PolyIOU loss: one quad-pair per thread. wave32, LDS transposed float2 layout.
//
// LDS map (per thread "column", float2 element e at smem[e*BLOCK + tid]):
//   e  0..3  : angle-sorted pred points   (kept intact for hull dynamic reads)
//   e  4..7  : angle-sorted target points (kept intact for hull dynamic reads)
//   e  8..15 : clip buffer A
//   e 16..23 : clip buffer B
// BLOCK=256 is a multiple of 64 banks -> lane t always hits bank pair
// (2t, 2t+1) mod 64; with 32 lanes/wave every b64 access is conflict-free
// for arbitrary divergent, data-dependent indices.

#define BLOCK 256
#define ELEMS 24

__global__ __launch_bounds__(BLOCK)
void poly_iou_loss_kernel(const float* __restrict__ preds,
                          const float* __restrict__ targets,
                          float* __restrict__ out, int n)
{
    __shared__ float2 smem[ELEMS * BLOCK];
    const int tid = threadIdx.x;
    const int gid = blockIdx.x * BLOCK + tid;
    if (gid >= n) return;

#define S2(e) smem[(e) * BLOCK + tid]

    // ---- coalesced 32B loads per input (global_load_b128) ----
    const float4* p4 = reinterpret_cast<const float4*>(preds) + (size_t)gid * 2;
    const float4* t4 = reinterpret_cast<const float4*>(targets) + (size_t)gid * 2;
    float4 pa = p4[0], pb = p4[1];
    float4 ta = t4[0], tb = t4[1];

    // ---- stable angle-sort of 4 points into LDS at 'base' (rank write is dynamic) ----
    auto sort4 = [&](float x0, float y0, float x1, float y1,
                     float x2, float y2, float x3, float y3, int base) {
        float cx = 0.25f * (x0 + x1 + x2 + x3);
        float cy = 0.25f * (y0 + y1 + y2 + y3);
        float a0 = atan2f(y0 - cy, x0 - cx);
        float a1 = atan2f(y1 - cy, x1 - cx);
        float a2 = atan2f(y2 - cy, x2 - cx);
        float a3 = atan2f(y3 - cy, x3 - cx);
        // stable ascending rank: r_i = #{j : a_j < a_i  or (a_j == a_i and j < i)}
        int r0 = (int)(a1 < a0) + (int)(a2 < a0) + (int)(a3 < a0);
        int r1 = (int)(a0 <= a1) + (int)(a2 < a1) + (int)(a3 < a1);
        int r2 = (int)(a0 <= a2) + (int)(a1 <= a2) + (int)(a3 < a2);
        int r3 = (int)(a0 <= a3) + (int)(a1 <= a3) + (int)(a2 <= a3);
        S2(base + r0) = make_float2(x0, y0);
        S2(base + r1) = make_float2(x1, y1);
        S2(base + r2) = make_float2(x2, y2);
        S2(base + r3) = make_float2(x3, y3);
    };
    sort4(pa.x, pa.y, pa.z, pa.w, pb.x, pb.y, pb.z, pb.w, 0); // sorted pred -> e0..3
    sort4(ta.x, ta.y, ta.z, ta.w, tb.x, tb.y, tb.z, tb.w, 4); // sorted tgt  -> e4..7

    // read sorted points back into registers once (static b64 loads)
    float2 P[4] = {S2(0), S2(1), S2(2), S2(3)};
    float2 T[4] = {S2(4), S2(5), S2(6), S2(7)};

    int n_poly;

    // ---- clip edge 0: subject is always the 4 sorted pred points (fully static) ----
    {
        float ex = T[1].x - T[0].x, ey = T[1].y - T[0].y;
        float ax = T[0].x, ay = T[0].y;
        float dP[4];
        #pragma unroll
        for (int i = 0; i < 4; ++i)
            dP[i] = ex * (P[i].y - ay) - ey * (P[i].x - ax);
        int cnt = 0;
        #pragma unroll
        for (int i = 0; i < 4; ++i) {
            const int j = (i + 1) & 3;
            bool in_cur = dP[i] >= 0.0f;
            bool in_nxt = dP[j] >= 0.0f;
            float den = dP[i] - dP[j];
            den = (fabsf(den) < 1e-12f) ? 1e-12f : den;
            float t = dP[i] / den;
            float2 ip;
            ip.x = P[i].x + t * (P[j].x - P[i].x);
            ip.y = P[i].y + t * (P[j].y - P[i].y);
            if (in_cur)            { S2(8 + cnt) = P[i]; ++cnt; }  // emission <= 6, no drop guard
            if (in_cur != in_nxt)  { S2(8 + cnt) = ip;   ++cnt; }
        }
        n_poly = cnt;
    }

    // ---- clip edges 1..3 (generic, with d-forwarding and i==7 jnp-clamp semantics) ----
    // schedule: edge1 8->16, edge2 16->8, edge3 8->16; trip bounds: 6, 8, 8
    const int srcs[3] = {8, 16, 8};
    const int dsts[3] = {16, 8, 16};
    const int lims[3] = {6, 8, 8};     // n_poly<=6 entering edge1 (<= floor(3*4/2))
    #pragma unroll
    for (int k = 0; k < 3; ++k) {
        const int src = srcs[k], dst = dsts[k], LIM = lims[k];
        float2 a = T[k + 1];
        float2 b = T[(k + 2) & 3];
        float ex = b.x - a.x, ey = b.y - a.y;
        int cnt = 0;
        float2 first = S2(src);
        float d_first = ex * (first.y - a.y) - ey * (first.x - a.x);
        float2 cur = first; float d_cur = d_first;
        #pragma unroll
        for (int i = 0; i < 8; ++i) {
            if (i >= LIM) break;       // folds at compile time
            if (i < n_poly) {
                bool last = (i == n_poly - 1);
                float2 nb; float d_nb;
                if (i < LIM - 1) {     // folds: load the next stored vertex
                    nb = S2(src + i + 1);
                    d_nb = ex * (nb.y - a.y) - ey * (nb.x - a.x);
                } else {               // i==7 and !last only if n_poly>8: jnp clamps to poly[7]
                    nb = cur; d_nb = d_cur;
                }
                float2 nxt;
                nxt.x = last ? first.x : nb.x;
                nxt.y = last ? first.y : nb.y;
                float d_nxt = last ? d_first : d_nb;

                bool in_cur = d_cur >= 0.0f;
                bool in_nxt = d_nxt >= 0.0f;
                float den = d_cur - d_nxt;
                den = (fabsf(den) < 1e-12f) ? 1e-12f : den;
                float t = d_cur / den;
                float2 ip;
                ip.x = cur.x + t * (nxt.x - cur.x);
                ip.y = cur.y + t * (nxt.y - cur.y);

                if (in_cur)           { if (cnt < 8) S2(dst + cnt) = cur; ++cnt; }
                if (in_cur != in_nxt) { if (cnt < 8) S2(dst + cnt) = ip;  ++cnt; }
                cur = nb; d_cur = d_nb;   // forward vertex and its distance
            }
        }
        n_poly = cnt;
    }

    // ---- intersection area (shoelace, final buffer = e16..23, same clamp semantics) ----
    float s = 0.0f;
    {
        float2 first = S2(16);
        float2 cur = first;
        #pragma unroll
        for (int i = 0; i < 8; ++i) {
            if (i < n_poly) {
                bool last = (i == n_poly - 1);
                float2 nb = (i < 7) ? S2(16 + i + 1) : cur;  // i==7,!last -> poly[7] (zero cross)
                float nxx = last ? first.x : nb.x;
                float nxy = last ? first.y : nb.y;
                s += cur.x * nxy - nxx * cur.y;
                cur = nb;
            }
        }
    }
    float inter = 0.5f * fabsf(s);

    // ---- convex hull area of the 8 combined sorted points (gift wrapping) ----
    float X[8], Y[8];
    #pragma unroll
    for (int i = 0; i < 4; ++i) { X[i] = P[i].x; Y[i] = P[i].y; }
    #pragma unroll
    for (int i = 0; i < 4; ++i) { X[4 + i] = T[i].x; Y[4 + i] = T[i].y; }

    int start = 0; float miny = Y[0];
    #pragma unroll
    for (int i = 1; i < 8; ++i) {
        if (Y[i] < miny) { miny = Y[i]; start = i; }  // first-occurrence argmin
    }

    int cur = start; bool done = false;
    float2 sp = S2(start);                 // dynamic, conflict-free (pts region intact)
    float px = sp.x, py = sp.y;
    float acc = 0.0f;

    #pragma unroll
    for (int step = 0; step < 8; ++step) {
        float rx[8], ry[8], d2[8]; bool v[8];
        #pragma unroll
        for (int i = 0; i < 8; ++i) {
            rx[i] = X[i] - px; ry[i] = Y[i] - py;
            d2[i] = rx[i] * rx[i] + ry[i] * ry[i];
            v[i] = d2[i] > 1e-12f;         // diagonal C_ii == 0 always passes
        }
        // antisymmetric cross matrix: only the 28-element upper triangle
        #pragma unroll
        for (int i = 0; i < 8; ++i) {
            #pragma unroll
            for (int j = i + 1; j < 8; ++j) {
                float C = rx[i] * ry[j] - ry[i] * rx[j];
                v[i] = v[i] && (C >= -1e-6f);   //  C_ij >= -eps
                v[j] = v[j] && (C <=  1e-6f);   // -C_ij >= -eps
            }
        }
        int nxt = 0; float best = v[0] ? d2[0] : -1.0f;
        #pragma unroll
        for (int i = 1; i < 8; ++i) {
            float score = v[i] ? d2[i] : -1.0f;
            if (score > best) { best = score; nxt = i; }  // first-occurrence argmax
        }
        if (done) nxt = cur;
        done = done || (nxt == start);
        float2 q = S2(nxt);                // dynamic, conflict-free
        acc += px * q.y - q.x * py;
        px = q.x; py = q.y; cur = nxt;
    }
    float hull = 0.5f * fabsf(acc + (px * sp.y - sp.x * py)); // closing edge

    float iou = (hull > 1e-12f) ? (inter / fmaxf(hull, 1e-12f)) : 0.0f;
    out[gid] = 1.0f - iou;
#undef S2
}

extern "C" void kernel_launch(void* const* d_in, const int* in_sizes, int n_in,
                              void* d_out, int out_size, void* d_ws, size_t ws_size,
                              hipStream_t stream) {
    const float* preds   = (const float*)d_in[0];
    const float* targets = (const float*)d_in[1];
    float* out = (float*)d_out;
    int n = in_sizes[0] / 8; // N pairs, 8 floats each
    int grid = (n + BLOCK - 1) / BLOCK;
    poly_iou_loss_kernel<<<grid, BLOCK, 0, stream>>>(preds, targets, out, n);
}